// YoloV3Loss_16776142258556
// MI455X (gfx1250) — compile-verified
//
#include <hip/hip_runtime.h>
#include <hip/hip_bf16.h>
#include <math.h>

// ---------------------------------------------------------------------------
// YOLOv3 loss, channel-sparse formulation for MI455X (gfx1250, wave32).
// Dense work = conf channels only (~1 MB); box/class terms read sparsely at
// <=1600 masked cells. WMMA f32 16x16x4 used as an exact cross-lane reducer.
// ---------------------------------------------------------------------------

typedef __attribute__((ext_vector_type(2))) float v2f;
typedef __attribute__((ext_vector_type(8))) float v8f;

#define WDIM 52
#define HW2704 (WDIM * WDIM)

struct Entry {
  int cell;
  int cid;
  float tx, ty, tw, th;
};

// min(softplus(z), 100) == -max(log(1-sigmoid(z)), -100), computed stably.
__device__ __forceinline__ float softplus_clamped(float z) {
  float az = fabsf(z);
  float sp = log1pf(__expf(-az)) + fmaxf(z, 0.0f);
  return fminf(sp, 100.0f);
}

// Exact 32-lane sum using V_WMMA_F32_16X16X4_F32 (full f32 precision).
// A: V0 = value, V1 = 0  => rowsum(m) = v[m] + v[m+16]   (A 16x4 layout)
// B: all ones (covers every B slot regardless of row mapping)
// D VGPR r, lane<16 holds rowsum(r); lane>=16 holds rowsum(r+8).
// Sum of the 8 D regs = half-wave sum; shfl_xor(16) folds the halves.
// EXEC must be all 1s at the call site (no divergent exits before this).
__device__ __forceinline__ float wave_sum32(float v) {
  v2f a;
  a[0] = v;
  a[1] = 0.0f;
  v2f b;
  b[0] = 1.0f;
  b[1] = 1.0f;
  v8f c = {};
  c = __builtin_amdgcn_wmma_f32_16x16x4_f32(
      /*neg_a=*/false, a, /*neg_b=*/false, b,
      /*c_mod=*/(short)0, c, /*reuse_a=*/false, /*reuse_b=*/false);
  float s = c[0] + c[1] + c[2] + c[3] + c[4] + c[5] + c[6] + c[7];
  s += __shfl_xor(s, 16, 32);
  return s;  // full 32-lane sum in every lane
}

// ---------------------------------------------------------------------------
__global__ void k_init(int* __restrict__ owner, unsigned char* __restrict__ noobj,
                       int* __restrict__ counter, float* __restrict__ accum, int NC) {
  int i = blockIdx.x * blockDim.x + threadIdx.x;
  int stride = gridDim.x * blockDim.x;
  for (int c = i; c < NC; c += stride) {
    owner[c] = -1;
    noobj[c] = 1;
  }
  if (i == 0) {
    *counter = 0;
    accum[0] = 0.0f;  // obj + noobj BCE sum
    accum[1] = 0.0f;  // n_pos
    accum[2] = 0.0f;  // box squared-error sum
    accum[3] = 0.0f;  // class BCE sum
  }
}

// ---------------------------------------------------------------------------
__global__ void k_encode(const float* __restrict__ tgt, int* __restrict__ owner,
                         unsigned char* __restrict__ noobj, Entry* __restrict__ entries,
                         int* __restrict__ counter, int B, int T) {
  int e = blockIdx.x * blockDim.x + threadIdx.x;
  if (e >= B * T) return;
  int b = e / T;
  const float* t = tgt + (size_t)e * 5;
  float t0 = t[0], t1 = t[1], t2 = t[2], t3 = t[3], t4 = t[4];
  if (!(t0 + t1 + t2 + t3 + t4 > 0.0f)) return;  // valid test

  // scaled anchors = ANCHORS * 52/416 = ANCHORS / 8
  const float AW[3] = {1.25f, 2.0f, 4.125f};
  const float AH[3] = {1.625f, 3.75f, 2.875f};

  float gx = t0 * (float)WDIM, gy = t1 * (float)WDIM;
  float gw = t2 * (float)WDIM, gh = t3 * (float)WDIM;
  int gi = (int)gx, gj = (int)gy;  // truncation matches astype(int32)

  float best_iou = -1.0f;
  int best = 0;
  for (int a = 0; a < 3; ++a) {
    float inter = fminf(gw, AW[a]) * fminf(gh, AH[a]);
    float iou = inter / (gw * gh + AW[a] * AH[a] - inter + 1e-16f);
    if (iou > best_iou) {
      best_iou = iou;
      best = a;
    }
    if (iou > 0.7f) {  // ignore-scatter: zero noobj at this anchor cell
      int cell = ((b * 3 + a) * WDIM + gj) * WDIM + gi;
      noobj[cell] = 0;
    }
  }

  int cell = ((b * 3 + best) * WDIM + gj) * WDIM + gi;
  int idx = atomicAdd(counter, 1);
  Entry en;
  en.cell = cell;
  en.cid = (int)t4;
  en.tx = gx - (float)gi;
  en.ty = gy - (float)gj;
  en.tw = gw / AW[best];
  en.th = gh / AH[best];
  entries[idx] = en;
  // last-writer-wins dedup of scatter collisions (mirrors .at[].set)
  atomicExch(&owner[cell], idx);
}

// ---------------------------------------------------------------------------
// One block per (b, anchor) conf plane: 2704 contiguous floats.
__global__ void k_dense(const float* __restrict__ inp, const int* __restrict__ owner,
                        const unsigned char* __restrict__ noobj, float* __restrict__ accum) {
  int p = blockIdx.x;  // b*3 + a
  int b = p / 3, a = p % 3;
  const float* cptr = inp + ((size_t)b * 255 + (size_t)a * 85 + 4) * HW2704;
  const int* own = owner + (size_t)p * HW2704;
  const unsigned char* nb = noobj + (size_t)p * HW2704;

  float s_obj = 0.0f;
  float s_np = 0.0f;
  for (int i = threadIdx.x; i < HW2704; i += blockDim.x) {
    __builtin_prefetch((const void*)(cptr + i + 256), 0, 1);  // global_prefetch_b8
    float z = cptr[i];
    if (own[i] >= 0) {  // masked cell: object_loss term = -log(sigmoid(z)) clamped
      s_obj += softplus_clamped(-z);
      s_np += 1.0f;
    }
    if (nb[i]) {  // noobj cell: -log(1 - sigmoid(z)) clamped
      s_obj += softplus_clamped(z);
    }
  }

  // Full EXEC here (no divergent returns above) -> WMMA legal.
  float tot_obj = wave_sum32(s_obj);
  float tot_np = wave_sum32(s_np);
  if ((threadIdx.x & 31) == 0) {
    atomicAdd(&accum[0], tot_obj);
    atomicAdd(&accum[1], tot_np);
  }
}

// ---------------------------------------------------------------------------
// One block per candidate entry: 80-class BCE + box MSE at the masked cell.
__global__ void k_sparse(const float* __restrict__ inp, const int* __restrict__ owner,
                         const Entry* __restrict__ entries, const int* __restrict__ counter,
                         float* __restrict__ accum) {
  __shared__ Entry se;
  __shared__ int go;
  __shared__ float red[128];

  int e = blockIdx.x;
  if (threadIdx.x == 0) {
    int ok = 0;
    if (e < *counter) {
      se = entries[e];
      if (owner[se.cell] == e) ok = 1;  // dedup: only the surviving writer
    }
    go = ok;
  }
  __syncthreads();
  if (!go) return;

  int cell = se.cell;
  int i = cell % WDIM;
  int j = (cell / WDIM) % WDIM;
  int a = (cell / HW2704) % 3;
  int b = cell / (3 * HW2704);
  size_t base = ((size_t)b * 255 + (size_t)a * 85) * HW2704 + (size_t)j * WDIM + i;

  float term = 0.0f;
  if (threadIdx.x < 80) {
    float z = inp[base + (size_t)(5 + threadIdx.x) * HW2704];
    // tcls one-hot: class cid -> -log(p); others -> -log(1-p), both clamped
    term = (threadIdx.x == se.cid) ? softplus_clamped(-z) : softplus_clamped(z);
  }
  red[threadIdx.x] = term;
  __syncthreads();
  for (int s = 64; s > 0; s >>= 1) {
    if (threadIdx.x < s) red[threadIdx.x] += red[threadIdx.x + s];
    __syncthreads();
  }

  if (threadIdx.x == 0) {
    float zx = inp[base + 0 * (size_t)HW2704];
    float zy = inp[base + 1 * (size_t)HW2704];
    float zw = inp[base + 2 * (size_t)HW2704];
    float zh = inp[base + 3 * (size_t)HW2704];
    float dx = 1.0f / (1.0f + __expf(-zx)) - se.tx;
    float dy = 1.0f / (1.0f + __expf(-zy)) - se.ty;
    float dw = __expf(zw) - se.tw;
    float dh = __expf(zh) - se.th;
    atomicAdd(&accum[2], dx * dx + dy * dy + dw * dw + dh * dh);
    atomicAdd(&accum[3], red[0]);
  }
}

// ---------------------------------------------------------------------------
__global__ void k_final(const float* __restrict__ accum, float* __restrict__ out, int B, int NC) {
  float N = (float)NC;
  float obj = accum[0];
  float npos = fmaxf(accum[1], 1.0f);
  float box = accum[2];
  float cls = accum[3];
  float loss = (5.0f * box / N + obj / N + cls / (npos * 80.0f)) * (float)B;
  out[0] = loss;
}

// ---------------------------------------------------------------------------
extern "C" void kernel_launch(void* const* d_in, const int* in_sizes, int n_in,
                              void* d_out, int out_size, void* d_ws, size_t ws_size,
                              hipStream_t stream) {
  const float* inp = (const float*)d_in[0];
  const float* tgt = (const float*)d_in[1];
  float* out = (float*)d_out;

  const int CH = 255;  // 3 * (5 + 80)
  int B = in_sizes[0] / (CH * HW2704);
  int T = in_sizes[1] / (B * 5);
  int NC = B * 3 * HW2704;

  // Workspace layout (~1.35 MB for B=32):
  char* ws = (char*)d_ws;
  int* owner = (int*)ws;                                    // NC ints
  unsigned char* noobj = (unsigned char*)(owner + NC);      // NC bytes
  size_t off = (size_t)NC * sizeof(int) + (size_t)NC;
  off = (off + 15) & ~(size_t)15;
  Entry* entries = (Entry*)(ws + off);                      // B*T entries
  off += sizeof(Entry) * (size_t)(B * T);
  off = (off + 15) & ~(size_t)15;
  int* counter = (int*)(ws + off);
  float* accum = (float*)(ws + off + 16);                   // 4 floats

  hipLaunchKernelGGL(k_init, dim3(256), dim3(256), 0, stream,
                     owner, noobj, counter, accum, NC);
  hipLaunchKernelGGL(k_encode, dim3((B * T + 255) / 256), dim3(256), 0, stream,
                     tgt, owner, noobj, entries, counter, B, T);
  hipLaunchKernelGGL(k_dense, dim3(B * 3), dim3(256), 0, stream,
                     inp, owner, noobj, accum);
  hipLaunchKernelGGL(k_sparse, dim3(B * T), dim3(128), 0, stream,
                     inp, owner, entries, counter, accum);
  hipLaunchKernelGGL(k_final, dim3(1), dim3(1), 0, stream, accum, out, B, NC);
}